// GCN_36344013259390
// MI455X (gfx1250) — compile-verified
//
#include <hip/hip_runtime.h>
#include <hip/hip_bf16.h>

typedef __attribute__((ext_vector_type(2))) float v2f;
typedef __attribute__((ext_vector_type(8))) float v8f;

#define IN_C  512
#define HID_C 64
#define OUT_C 40

// ---------------- elementwise helpers ----------------

__global__ void k_fill(float* __restrict__ p, float v, long long n) {
    long long i = (long long)blockIdx.x * blockDim.x + threadIdx.x;
    if (i < n) p[i] = v;
}

__global__ void k_deg(const long long* __restrict__ ei, long long E,
                      float* __restrict__ deg) {
    long long e = (long long)blockIdx.x * blockDim.x + threadIdx.x;
    if (e < E) {
        long long dst = ei[E + e];
        unsafeAtomicAdd(&deg[dst], 1.0f);
    }
}

__global__ void k_rsqrt(float* __restrict__ deg, int n) {
    int i = blockIdx.x * blockDim.x + threadIdx.x;
    if (i < n) deg[i] = rsqrtf(deg[i]);
}

// edge scatter: out[dst] += dinv[src]*dinv[dst] * h[src]
// 16 lanes / edge, float4 gather, 4 hardware float atomics per lane
template <int C>
__global__ __launch_bounds__(256) void k_scatter(const float* __restrict__ h,
                                                 const float* __restrict__ dinv,
                                                 const long long* __restrict__ ei,
                                                 long long E, float* __restrict__ out) {
    const int c4 = (threadIdx.x & 15) * 4;
    long long e = (long long)blockIdx.x * 16 + (threadIdx.x >> 4);
    if (e < E && c4 < C) {
        long long s = ei[e];
        long long d = ei[E + e];
        float w = dinv[s] * dinv[d];
        const float4 hv = *(const float4*)(h + s * C + c4);
        float* po = out + d * C + c4;
        unsafeAtomicAdd(po + 0, w * hv.x);
        unsafeAtomicAdd(po + 1, w * hv.y);
        unsafeAtomicAdd(po + 2, w * hv.z);
        unsafeAtomicAdd(po + 3, w * hv.w);
    }
}

// ---------------- WMMA fp32 GEMMs ----------------
// V_WMMA_F32_16X16X4_F32 fragment mapping (ISA 7.12.2):
//   A(16x4):  lane&15 = M row; half=lane>>4 selects K base (0 or 2); v[0]=K, v[1]=K+1
//   B(4x16):  lane&15 = N col; same K split
//   C(16x16): VGPR r -> M = r + 8*half, N = lane&15

// GEMM1: h1 = x @ W1, fused epilogue also writes agg1 = b1 + dinv^2 * h1
__global__ __launch_bounds__(128) void k_gemm1(const float* __restrict__ X,
                                               const float* __restrict__ W,
                                               const float* __restrict__ B1,
                                               const float* __restrict__ dinv,
                                               float* __restrict__ H,
                                               float* __restrict__ AGG, int n) {
    const int lane    = threadIdx.x & 31;
    const int half    = lane >> 4;
    const int m15     = lane & 15;
    const int colBase = (threadIdx.x >> 5) * 16;   // 4 waves -> cols 0,16,32,48
    const int rowBase = blockIdx.x * 16;

    int row = rowBase + m15;
    if (row >= n) row = n - 1;                     // clamp loads; stores masked in tail
    const float* Ar = X + (size_t)row * IN_C;
    const int col  = colBase + m15;
    const float bias = B1[col];

    v8f acc = {};
    for (int k = 0; k < IN_C; k += 4) {
        const int kk = k + 2 * half;
        v2f a, b;
        a.x = Ar[kk];
        a.y = Ar[kk + 1];
        b.x = W[(size_t)kk * HID_C + col];
        b.y = W[(size_t)(kk + 1) * HID_C + col];
        acc = __builtin_amdgcn_wmma_f32_16x16x4_f32(false, a, false, b,
                                                    (short)0, acc, false, false);
    }

    if (rowBase + 16 <= n) {                       // uniform fast path, no masking
#pragma unroll
        for (int r = 0; r < 8; ++r) {
            const int orow = rowBase + r + 8 * half;
            const size_t idx = (size_t)orow * HID_C + col;
            const float v = acc[r];
            const float dd = dinv[orow];
            H[idx]   = v;
            AGG[idx] = bias + dd * dd * v;
        }
    } else {
#pragma unroll
        for (int r = 0; r < 8; ++r) {
            const int orow = rowBase + r + 8 * half;
            if (orow < n) {
                const size_t idx = (size_t)orow * HID_C + col;
                const float v = acc[r];
                const float dd = dinv[orow];
                H[idx]   = v;
                AGG[idx] = bias + dd * dd * v;
            }
        }
    }
}

// GEMM2: h2 = relu(agg1) @ W2, fused epilogue also writes out = b2 + dinv^2 * h2
__global__ __launch_bounds__(96) void k_gemm2(const float* __restrict__ A,
                                              const float* __restrict__ W,
                                              const float* __restrict__ B2,
                                              const float* __restrict__ dinv,
                                              float* __restrict__ H,
                                              float* __restrict__ OUT, int n) {
    const int lane    = threadIdx.x & 31;
    const int half    = lane >> 4;
    const int m15     = lane & 15;
    const int colBase = (threadIdx.x >> 5) * 16;   // 3 waves -> cols 0,16,32
    const int rowBase = blockIdx.x * 16;

    int row = rowBase + m15;
    if (row >= n) row = n - 1;
    const float* Ar = A + (size_t)row * HID_C;
    const int col  = colBase + m15;
    const int bcol = (col < OUT_C) ? col : (OUT_C - 1);  // clamp W/bias loads
    const float bias = B2[bcol];

    v8f acc = {};
    for (int k = 0; k < HID_C; k += 4) {
        const int kk = k + 2 * half;
        v2f a, b;
        a.x = fmaxf(Ar[kk], 0.0f);                 // fused ReLU
        a.y = fmaxf(Ar[kk + 1], 0.0f);
        b.x = W[(size_t)kk * OUT_C + bcol];
        b.y = W[(size_t)(kk + 1) * OUT_C + bcol];
        acc = __builtin_amdgcn_wmma_f32_16x16x4_f32(false, a, false, b,
                                                    (short)0, acc, false, false);
    }

    if (col < OUT_C) {
        if (rowBase + 16 <= n) {                   // uniform fast path
#pragma unroll
            for (int r = 0; r < 8; ++r) {
                const int orow = rowBase + r + 8 * half;
                const size_t idx = (size_t)orow * OUT_C + col;
                const float v = acc[r];
                const float dd = dinv[orow];
                H[idx]   = v;
                OUT[idx] = bias + dd * dd * v;
            }
        } else {
#pragma unroll
            for (int r = 0; r < 8; ++r) {
                const int orow = rowBase + r + 8 * half;
                if (orow < n) {
                    const size_t idx = (size_t)orow * OUT_C + col;
                    const float v = acc[r];
                    const float dd = dinv[orow];
                    H[idx]   = v;
                    OUT[idx] = bias + dd * dd * v;
                }
            }
        }
    }
}

// ---------------- softmax: one wave32 per row of 40 ----------------
__global__ __launch_bounds__(256) void k_softmax(float* __restrict__ out, int n) {
    int lane = threadIdx.x & 31;
    int row  = blockIdx.x * 8 + (threadIdx.x >> 5);
    if (row >= n) return;                          // whole wave exits together
    float* p = out + (size_t)row * OUT_C;

    float v0 = p[lane];                            // cols 0..31
    bool hi  = lane < (OUT_C - 32);                // cols 32..39
    float v1 = hi ? p[32 + lane] : -INFINITY;

    float m = fmaxf(v0, v1);
    for (int off = 16; off; off >>= 1) m = fmaxf(m, __shfl_xor(m, off, 32));

    float e0 = __expf(v0 - m);
    float e1 = hi ? __expf(v1 - m) : 0.0f;
    float s  = e0 + e1;
    for (int off = 16; off; off >>= 1) s += __shfl_xor(s, off, 32);

    float inv = 1.0f / s;
    p[lane] = e0 * inv;
    if (hi) p[32 + lane] = e1 * inv;
}

// ---------------- launcher ----------------
extern "C" void kernel_launch(void* const* d_in, const int* in_sizes, int n_in,
                              void* d_out, int out_size, void* d_ws, size_t ws_size,
                              hipStream_t stream) {
    const float*     x  = (const float*)d_in[0];
    const long long* ei = (const long long*)d_in[1];
    const float*     W1 = (const float*)d_in[2];
    const float*     b1 = (const float*)d_in[3];
    const float*     W2 = (const float*)d_in[4];
    const float*     b2 = (const float*)d_in[5];
    float* out = (float*)d_out;

    const int       n = in_sizes[0] / IN_C;   // 50000
    const long long E = in_sizes[1] / 2;      // 800000

    // workspace carve-out (256B aligned): dinv | h1 | agg1 | h2  (~34 MB)
    char*  wsb = (char*)d_ws;
    size_t off = 0;
    auto carve = [&](size_t bytes) -> void* {
        void* p = wsb + off;
        off = (off + bytes + 255) & ~(size_t)255;
        return p;
    };
    float* dinv = (float*)carve((size_t)n * sizeof(float));
    float* h1   = (float*)carve((size_t)n * HID_C * sizeof(float));
    float* agg1 = (float*)carve((size_t)n * HID_C * sizeof(float));
    float* h2   = (float*)carve((size_t)n * OUT_C * sizeof(float));

    const int rowTiles = (n + 15) / 16;

    // degrees (self-loop contributes 1) -> dinv = deg^-1/2
    k_fill<<<(unsigned)((n + 255) / 256), 256, 0, stream>>>(dinv, 1.0f, (long long)n);
    k_deg<<<(unsigned)((E + 255) / 256), 256, 0, stream>>>(ei, E, dinv);
    k_rsqrt<<<(unsigned)((n + 255) / 256), 256, 0, stream>>>(dinv, n);

    // layer 1: GEMM (+ fused bias/self-loop seed), then edge scatter
    k_gemm1<<<rowTiles, 128, 0, stream>>>(x, W1, b1, dinv, h1, agg1, n);
    k_scatter<HID_C><<<(unsigned)((E + 15) / 16), 256, 0, stream>>>(h1, dinv, ei, E, agg1);

    // layer 2: GEMM with fused ReLU on loads (+ fused bias/self-loop seed), scatter
    k_gemm2<<<rowTiles, 96, 0, stream>>>(agg1, W2, b2, dinv, h2, out, n);
    k_scatter<OUT_C><<<(unsigned)((E + 15) / 16), 256, 0, stream>>>(h2, dinv, ei, E, out);

    // row softmax
    k_softmax<<<(unsigned)((n + 7) / 8), 256, 0, stream>>>(out, n);
}